// HierarchicalReconstructionModule_45569603010587
// MI455X (gfx1250) — compile-verified
//
#include <hip/hip_runtime.h>
#include <hip/hip_bf16.h>

typedef __attribute__((ext_vector_type(2))) float v2f;
typedef __attribute__((ext_vector_type(8))) float v8f;

#define N_BEADS 1000
#define A_ATOMS 12

// One wave (32 lanes) per 16-bead tile.
// P = RV * T^T + bead_pos  via 3 chained V_WMMA_F32_16X16X4_F32 per coordinate,
// where T is the constant lower-triangular prefix mask (k>=1).
__global__ __launch_bounds__(32)
void hier_recon_wmma(const float* __restrict__ relvec,       // (N, A, 3)
                     const float* __restrict__ bead_pos,     // (N, 3)
                     const float* __restrict__ weights,      // (N, A)
                     const float* __restrict__ bond_lengths, // (21, A, 1)
                     const int*   __restrict__ bead_types,   // (N,)
                     float*       __restrict__ out)          // (N*A, 3)
{
    const int lane = threadIdx.x;     // wave32
    const int half = lane >> 4;       // 0 | 1
    const int col  = lane & 15;       // B/C/D column (atom index), A-matrix row
    const int b0   = blockIdx.x * 16; // first bead of this tile

    // ---- A-matrix operand: rv[bead = b0+col, atom = 4q + 2*half + j, coord]
    int beadA = b0 + col; if (beadA >= N_BEADS) beadA = N_BEADS - 1;
    const int typeA = bead_types[beadA];

    float aval[3][2][3];
#pragma unroll
    for (int q = 0; q < 3; ++q) {
#pragma unroll
        for (int j = 0; j < 2; ++j) {
            const int atom = 4*q + 2*half + j;
            const float* rp = relvec + ((size_t)beadA * A_ATOMS + atom) * 3;
            const float x = rp[0], y = rp[1], z = rp[2];
            const float s = bond_lengths[typeA * A_ATOMS + atom] /
                            (sqrtf(x*x + y*y + z*z) + 1e-5f);
            aval[q][j][0] = x * s;
            aval[q][j][1] = y * s;
            aval[q][j][2] = z * s;
        }
    }

    // ---- B matrices: constant prefix masks, B[k_global, n] = (1 <= k_global <= n)
    //      layout: VGPR j holds K-row (j + 2*half), column = col
    v2f Bm[3];
#pragma unroll
    for (int q = 0; q < 3; ++q) {
#pragma unroll
        for (int j = 0; j < 2; ++j) {
            const int kg = 4*q + 2*half + j;
            Bm[q][j] = (kg >= 1 && kg <= col) ? 1.0f : 0.0f;
        }
    }

    // ---- per-row data in C/D layout: VGPR rr -> bead row M = rr + 8*half
    float bp[8][3]; float w8[8]; int rb[8];
#pragma unroll
    for (int rr = 0; rr < 8; ++rr) {
        const int bead = b0 + rr + 8*half;
        rb[rr] = bead;
        const int bc = (bead < N_BEADS) ? bead : N_BEADS - 1;
        bp[rr][0] = bead_pos[bc*3 + 0];
        bp[rr][1] = bead_pos[bc*3 + 1];
        bp[rr][2] = bead_pos[bc*3 + 2];
        w8[rr] = (col < A_ATOMS) ? weights[bc*A_ATOMS + col] : 0.0f;
    }

#pragma unroll
    for (int c = 0; c < 3; ++c) {
        // accumulator seeded with bead_pos broadcast along atoms
        v8f d;
#pragma unroll
        for (int rr = 0; rr < 8; ++rr) d[rr] = bp[rr][c];

        // prefix sum over 12 atoms = 3 chained 16x16x4 f32 WMMAs
#pragma unroll
        for (int q = 0; q < 3; ++q) {
            v2f Am; Am[0] = aval[q][0][c]; Am[1] = aval[q][1][c];
            d = __builtin_amdgcn_wmma_f32_16x16x4_f32(
                    /*neg_a=*/false, Am, /*neg_b=*/false, Bm[q],
                    /*c_mod=*/(short)0, d, /*reuse_a=*/false, /*reuse_b=*/false);
        }

        // shift = sum_a w*p - bead_pos ; out = p - shift
#pragma unroll
        for (int rr = 0; rr < 8; ++rr) {
            float t = w8[rr] * d[rr];
            t += __shfl_xor(t, 1, 32);
            t += __shfl_xor(t, 2, 32);
            t += __shfl_xor(t, 4, 32);
            t += __shfl_xor(t, 8, 32);          // row sum over the 16 columns of this half
            const float shift = t - bp[rr][c];
            if (col < A_ATOMS && rb[rr] < N_BEADS)
                out[((size_t)rb[rr]*A_ATOMS + col)*3 + c] = d[rr] - shift;
        }
    }
}

extern "C" void kernel_launch(void* const* d_in, const int* in_sizes, int n_in,
                              void* d_out, int out_size, void* d_ws, size_t ws_size,
                              hipStream_t stream) {
    const float* relvec       = (const float*)d_in[0]; // (N, A, 3)
    const float* bead_pos     = (const float*)d_in[1]; // (N, 3)
    const float* weights      = (const float*)d_in[2]; // (N, A)
    const float* bond_lengths = (const float*)d_in[3]; // (21, A, 1)
    const int*   bead_types   = (const int*)d_in[4];   // (N,)
    float* out = (float*)d_out;                        // (N*A, 3)

    const int nwaves = (N_BEADS + 15) / 16;            // 63
    hier_recon_wmma<<<nwaves, 32, 0, stream>>>(relvec, bead_pos, weights,
                                               bond_lengths, bead_types, out);
}